// Generator_50989851738492
// MI455X (gfx1250) — compile-verified
//
#include <hip/hip_runtime.h>

typedef __attribute__((ext_vector_type(2))) float v2f;
typedef __attribute__((ext_vector_type(8))) float v8f;

#define BB  8
#define CC  16
#define DD  192
#define OCC 4
#define OHH 50
#define OWW 200
#define TLD 194   // padded LDS leading dim (192 % 64 == 0 would serialize banks)
#define JG  4     // j-split groups for the projection kernel

__device__ __forceinline__ v8f vzero8() {
  v8f v;
#pragma unroll
  for (int k = 0; k < 8; ++k) v[k] = 0.0f;
  return v;
}

__device__ __forceinline__ v8f wmma_f32(v2f a, v2f b, v8f c) {
  // V_WMMA_F32_16X16X4_F32: D = A(16x4) * B(4x16) + C(16x16), fp32 throughout
  return __builtin_amdgcn_wmma_f32_16x16x4_f32(
      /*neg_a=*/false, a, /*neg_b=*/false, b,
      /*c_mod=*/(short)0, c, /*reuse_a=*/false, /*reuse_b=*/false);
}

// A-operand fragment: 16x4 tile of a row-major matrix.
// lanes 0-15: row m0+lane, K = k0+0,k0+1 ; lanes 16-31: row m0+lane-16, K = k0+2,k0+3
__device__ __forceinline__ v2f frag_row(const float* base, int ld, int row0, int k0, int lane) {
  int r = row0 + (lane & 15);
  int k = k0 + ((lane >> 4) << 1);
  const float* p = base + r * ld + k;
  v2f f;
  f.x = p[0];
  f.y = p[1];
  return f;
}

__device__ __forceinline__ v2f frag_row_clamp(const float* base, int ld, int row0, int k0,
                                              int lane, int maxrow) {
  int r = row0 + (lane & 15);
  if (r > maxrow) r = maxrow;
  int k = k0 + ((lane >> 4) << 1);
  const float* p = base + r * ld + k;
  v2f f;
  f.x = p[0];
  f.y = p[1];
  return f;
}

// B-operand fragment: 4x16 tile of a row-major KxN matrix.
// lanes 0-15: col n0+lane, K rows k0+0,k0+1 ; lanes 16-31: col n0+lane-16, K rows k0+2,k0+3
__device__ __forceinline__ v2f frag_col(const float* base, int ld, int k0, int col0, int lane) {
  int n = col0 + (lane & 15);
  int k = k0 + ((lane >> 4) << 1);
  const float* p = base + k * ld + n;
  v2f f;
  f.x = p[0];
  f.y = p[ld];
  return f;
}

__device__ __forceinline__ v2f frag_col_clamp(const float* base, int ld, int k0, int col0,
                                              int lane, int maxcol) {
  int n = col0 + (lane & 15);
  if (n > maxcol) n = maxcol;
  int k = k0 + ((lane >> 4) << 1);
  const float* p = base + k * ld + n;
  v2f f;
  f.x = p[0];
  f.y = p[ld];
  return f;
}

// C/D layout: VGPR r -> lanes 0-15 row m0+r, lanes 16-31 row m0+8+r, col = n0 + lane%16
__device__ __forceinline__ void tile_store(float* base, int ld, int m0, int n0, v8f acc, int lane) {
  int cn = lane & 15;
  int mb = m0 + ((lane >> 4) << 3);
#pragma unroll
  for (int r = 0; r < 8; ++r) base[(mb + r) * ld + n0 + cn] = acc[r];
}

// ---------------------------------------------------------------------------
// Kernel 1: Y[b,i] = sum_j R[i,j] @ X[b,j] @ R[i,j]^T        (192x192 each)
// One block per (b, i, row-half); 8 waves; each wave owns a 48x48 (3x3 tile)
// patch of a 96x192 half. The half-rows temporary T = (R@X)[rows] lives in
// LDS (96 x TLD = 74.5 KB -> four workgroups fit per WGP's 320 KB).
// ---------------------------------------------------------------------------
__global__ void __launch_bounds__(256) rot_kernel(const float* __restrict__ x,
                                                  const float* __restrict__ R,
                                                  float* __restrict__ y) {
  extern __shared__ float Tl[];  // 96 * TLD floats, local rows 0..95
  const int rh   = blockIdx.x & 1;        // row half
  const int bi   = blockIdx.x >> 1;
  const int b    = bi / CC;
  const int i    = bi % CC;
  const int lane = threadIdx.x & 31;
  const int wid  = threadIdx.x >> 5;      // 0..7
  const int wr   = wid >> 2;              // 0..1 -> local rows 48*wr
  const int wc   = wid & 3;               // 0..3 -> cols 48*wc
  const int mg0  = 96 * rh + 48 * wr;     // global row base of this wave
  const int ml0  = 48 * wr;               // local (LDS) row base

  v8f yacc[3][3];
#pragma unroll
  for (int a = 0; a < 3; ++a)
#pragma unroll
    for (int c = 0; c < 3; ++c) yacc[a][c] = vzero8();

  for (int j = 0; j < CC; ++j) {
    const float* Rij = R + (i * CC + j) * DD * DD;
    const float* Xbj = x + (b * CC + j) * DD * DD;

    // ---- Phase 1: T[rows half] = R_ij[rows half] @ X_bj  -> LDS ----
    v8f t[3][3];
#pragma unroll
    for (int a = 0; a < 3; ++a)
#pragma unroll
      for (int c = 0; c < 3; ++c) t[a][c] = vzero8();

    for (int k0 = 0; k0 < DD; k0 += 4) {
      v2f af[3], bf[3];
#pragma unroll
      for (int a = 0; a < 3; ++a) af[a] = frag_row(Rij, DD, mg0 + 16 * a, k0, lane);
#pragma unroll
      for (int c = 0; c < 3; ++c) bf[c] = frag_col(Xbj, DD, k0, 48 * wc + 16 * c, lane);
#pragma unroll
      for (int a = 0; a < 3; ++a)
#pragma unroll
        for (int c = 0; c < 3; ++c) t[a][c] = wmma_f32(af[a], bf[c], t[a][c]);
    }
#pragma unroll
    for (int a = 0; a < 3; ++a)
#pragma unroll
      for (int c = 0; c < 3; ++c)
        tile_store(Tl, TLD, ml0 + 16 * a, 48 * wc + 16 * c, t[a][c], lane);
    __syncthreads();

    // ---- Phase 2: Yacc += T @ R_ij^T ----
    // B[q][n] = R[n][q]  => B fragment is a row-per-lane read of R rows n
    for (int q0 = 0; q0 < DD; q0 += 4) {
      v2f af[3], bf[3];
#pragma unroll
      for (int a = 0; a < 3; ++a) af[a] = frag_row(Tl, TLD, ml0 + 16 * a, q0, lane);
#pragma unroll
      for (int c = 0; c < 3; ++c) bf[c] = frag_row(Rij, DD, 48 * wc + 16 * c, q0, lane);
#pragma unroll
      for (int a = 0; a < 3; ++a)
#pragma unroll
        for (int c = 0; c < 3; ++c) yacc[a][c] = wmma_f32(af[a], bf[c], yacc[a][c]);
    }
    __syncthreads();  // protect Tl before next j overwrites it
  }

  float* Ybi = y + (b * CC + i) * DD * DD;
#pragma unroll
  for (int a = 0; a < 3; ++a)
#pragma unroll
    for (int c = 0; c < 3; ++c)
      tile_store(Ybi, DD, mg0 + 16 * a, 48 * wc + 16 * c, yacc[a][c], lane);
}

// ---------------------------------------------------------------------------
// Kernel 2: partial[g][b,i] = sum_{j in group g} P[i,j] @ Y[b,j] @ PT[i,j]
// One block per (b, i in OC, j-group); 8 waves. Rows padded 50->64 (clamped
// loads), cols 200 -> 13 tiles (clamped loads, masked stores). Each
// wave-group of 4 handles 7 column tiles (tile 6 duplicated -> full-EXEC
// WMMA, identical duplicate stores).
// ---------------------------------------------------------------------------
__global__ void __launch_bounds__(256) proj_kernel(const float* __restrict__ y,
                                                   const float* __restrict__ P,
                                                   const float* __restrict__ PT,
                                                   float* __restrict__ part) {
  extern __shared__ float Tl[];  // 64 * TLD floats
  const int g    = blockIdx.x & (JG - 1);
  const int bi   = blockIdx.x >> 2;       // JG == 4
  const int b    = bi / OCC;
  const int i    = bi % OCC;
  const int lane = threadIdx.x & 31;
  const int wid  = threadIdx.x >> 5;      // 0..7
  const int tr   = wid & 3;               // tile row -> rows 16*tr
  const int cg   = wid >> 2;              // 0..1 column group
  const int m0   = 16 * tr;

  v8f oacc[7];
#pragma unroll
  for (int c = 0; c < 7; ++c) oacc[c] = vzero8();

  for (int jj = 0; jj < CC / JG; ++jj) {
    const int j = g * (CC / JG) + jj;
    const float* Pij  = P + (i * CC + j) * OHH * DD;
    const float* PTij = PT + (i * CC + j) * DD * OWW;
    const float* Ybj  = y + (b * CC + j) * DD * DD;

    // ---- Phase 1: T2(64x192) = P_ij(padded) @ Y_bj ; wave: row tr, cols 6*cg..6*cg+5 ----
    v8f t[6];
#pragma unroll
    for (int c = 0; c < 6; ++c) t[c] = vzero8();
    for (int k0 = 0; k0 < DD; k0 += 4) {
      v2f af = frag_row_clamp(Pij, DD, m0, k0, lane, OHH - 1);
#pragma unroll
      for (int c = 0; c < 6; ++c) {
        v2f bf = frag_col(Ybj, DD, k0, 16 * (6 * cg + c), lane);
        t[c] = wmma_f32(af, bf, t[c]);
      }
    }
#pragma unroll
    for (int c = 0; c < 6; ++c) tile_store(Tl, TLD, m0, 16 * (6 * cg + c), t[c], lane);
    __syncthreads();

    // ---- Phase 2: oacc += T2 @ PT_ij ; wave: row tr, col tiles 6*cg + 0..6 ----
    for (int q0 = 0; q0 < DD; q0 += 4) {
      v2f af = frag_row(Tl, TLD, m0, q0, lane);
#pragma unroll
      for (int c = 0; c < 7; ++c) {
        v2f bf = frag_col_clamp(PTij, OWW, q0, 16 * (6 * cg + c), lane, OWW - 1);
        oacc[c] = wmma_f32(af, bf, oacc[c]);
      }
    }
    __syncthreads();
  }

  float* Pout  = part + ((size_t)g * BB * OCC + b * OCC + i) * OHH * OWW;
  const int cn = lane & 15;
  const int mb = m0 + ((lane >> 4) << 3);
#pragma unroll
  for (int c = 0; c < 7; ++c) {
    int n = 16 * (6 * cg + c) + cn;
#pragma unroll
    for (int r = 0; r < 8; ++r) {
      int m = mb + r;
      if (m < OHH && n < OWW) Pout[m * OWW + n] = oacc[c][r];
    }
  }
}

// Deterministic fixed-order 4-way sum of the j-group partials.
__global__ void __launch_bounds__(256) reduce_kernel(const float* __restrict__ part,
                                                     float* __restrict__ out, int n) {
  int idx = blockIdx.x * 256 + threadIdx.x;
  if (idx < n) {
    float s = part[idx];
#pragma unroll
    for (int g = 1; g < JG; ++g) s += part[(size_t)g * n + idx];
    out[idx] = s;
  }
}

extern "C" void kernel_launch(void* const* d_in, const int* in_sizes, int n_in,
                              void* d_out, int out_size, void* d_ws, size_t ws_size,
                              hipStream_t stream) {
  const float* x  = (const float*)d_in[0];   // [B,C,D,D]
  const float* R  = (const float*)d_in[1];   // [C,C,D,D]
  const float* P  = (const float*)d_in[2];   // [OC,C,OH,D]
  const float* PT = (const float*)d_in[3];   // [OC,C,D,OW]
  float* out = (float*)d_out;                // [B,OC,OH,OW]

  float* yws  = (float*)d_ws;                          // Y: B*C*D*D floats (18.9 MB)
  float* pws  = yws + (size_t)BB * CC * DD * DD;       // partials: JG*B*OC*OH*OW (5.1 MB)
  const int nout = BB * OCC * OHH * OWW;               // 320,000

  const uint32_t lds1 = 96 * TLD * sizeof(float);      // 74,496 B
  const uint32_t lds2 = 64 * TLD * sizeof(float);      // 49,664 B
  // Allow >64KB dynamic LDS (gfx1250 WGP has 320KB); idempotent, capture-safe.
  (void)hipFuncSetAttribute((const void*)rot_kernel,
                            hipFuncAttributeMaxDynamicSharedMemorySize, (int)lds1);

  rot_kernel<<<dim3(BB * CC * 2), dim3(256), lds1, stream>>>(x, R, yws);
  proj_kernel<<<dim3(BB * OCC * JG), dim3(256), lds2, stream>>>(yws, P, PT, pws);
  reduce_kernel<<<dim3((nout + 255) / 256), dim3(256), 0, stream>>>(pws, out, nout);
}